// FlyLoRALinear_32203664786073
// MI455X (gfx1250) — compile-verified
//
#include <hip/hip_runtime.h>

// FlyLoRA linear for MI455X (gfx1250, wave32, WMMA).
// x:[4,4096,4096]f32, A:[32,4096]f32, B:[4096,32]f32, d:[32]f32 -> out:[4,4096,4096]f32
// y = x A^T ; keep top-8 of |y+d| per token ; out = (y*mask) B^T * (64/32)

typedef __attribute__((ext_vector_type(2))) float v2f;
typedef __attribute__((ext_vector_type(4))) float v4f;
typedef __attribute__((ext_vector_type(8))) float v8f;

#define T_TOT   16384   // 4 * 4096 tokens
#define D_DIM   4096
#define R_DIM   32
#define K_TOP   8
#define LSCALE  2.0f    // alpha / r
#define KCHUNK  (D_DIM / 8)   // 512: K-split across the 8 waves of a block
#define ROWP    33            // padded LDS row stride (bank-conflict free)

// ---------------------------------------------------------------------------
// Kernel 1: y = x @ A^T  (WMMA f32 16x16x4) with K split across 8 waves for
// 8x memory-level parallelism, deterministic LDS tree reduction, then
// per-token top-8 routing. One block = one 16-token tile.
// ---------------------------------------------------------------------------
__global__ __launch_bounds__(256) void fly_project_route(
    const float* __restrict__ x, const float* __restrict__ A,
    const float* __restrict__ d, float* __restrict__ z)
{
  __shared__ __align__(16) float part[8 * 16 * ROWP];  // per-wave partial tiles
  __shared__ __align__(16) float red[16 * ROWP];       // reduced y tile

  const int lane  = threadIdx.x & 31;
  const int wv    = threadIdx.x >> 5;                  // 0..7: K-chunk owner
  const int tok0  = blockIdx.x * 16;
  const int row   = lane & 15;                         // M (A-op) / N (B-op)
  const int khalf = (lane >> 4) * 2;                   // K sub-offset (ISA layout)
  const int k0    = wv * KCHUNK;

  const float* xrow  = x + (size_t)(tok0 + row) * D_DIM + k0 + khalf;
  const float* arow0 = A + (size_t)row        * D_DIM + k0 + khalf;
  const float* arow1 = A + (size_t)(row + 16) * D_DIM + k0 + khalf;

  v8f c0 = {};  // y[tok][rank 0..15]
  v8f c1 = {};  // y[tok][rank 16..31]

  #pragma unroll 8
  for (int k = 0; k < KCHUNK; k += 4) {
    v2f a  = __builtin_nontemporal_load((const v2f*)(xrow + k));  // 256MB stream: TH=NT
    v2f b0 = *(const v2f*)(arow0 + k);                            // A stays L2-resident
    v2f b1 = *(const v2f*)(arow1 + k);
    c0 = __builtin_amdgcn_wmma_f32_16x16x4_f32(false, a, false, b0, (short)0, c0, false, false);
    c1 = __builtin_amdgcn_wmma_f32_16x16x4_f32(false, a, false, b1, (short)0, c1, false, false);
  }

  // Scatter C/D tile to this wave's LDS slice: VGPR i -> token i / i+8.
  float* my = part + wv * (16 * ROWP);
  const int thalf = (lane >> 4) * 8;
  #pragma unroll
  for (int i = 0; i < 8; ++i) {
    my[(i + thalf) * ROWP + row]      = c0[i];
    my[(i + thalf) * ROWP + row + 16] = c1[i];
  }
  __syncthreads();

  // Deterministic 8-way reduction: each (tok,r) element owned by one thread,
  // fixed summation order -> bit-stable across replays.
  #pragma unroll
  for (int e = threadIdx.x; e < 512; e += 256) {
    const int tok = e >> 5, r = e & 31;
    float s = 0.0f;
    #pragma unroll
    for (int w = 0; w < 8; ++w) s += part[w * (16 * ROWP) + tok * ROWP + r];
    red[tok * ROWP + r] = s;
  }
  __syncthreads();

  // Routing: threads 0..15 each own one token; top-8 of |y + d| over r=0..31,
  // strict '>' => lowest index wins ties (matches jax.lax.top_k).
  if (threadIdx.x < 16) {
    const int t = threadIdx.x;
    float yv[R_DIM], sc[R_DIM];
    #pragma unroll
    for (int r = 0; r < R_DIM; ++r) {
      yv[r] = red[t * ROWP + r];
      sc[r] = fabsf(yv[r] + d[r]);
    }
    unsigned sel = 0u;
    for (int j = 0; j < K_TOP; ++j) {
      float best = -1.0f; int bi = 0;
      for (int r = 0; r < R_DIM; ++r) {
        if (!((sel >> r) & 1u) && sc[r] > best) { best = sc[r]; bi = r; }
      }
      sel |= 1u << bi;
    }
    // Fold alpha/r into z so kernel 2 is a plain GEMM.
    float* zr = z + (size_t)(tok0 + t) * R_DIM;
    #pragma unroll
    for (int r = 0; r < R_DIM; r += 4) {
      v4f v;
      v.x = ((sel >> (r+0)) & 1u) ? yv[r+0] * LSCALE : 0.0f;
      v.y = ((sel >> (r+1)) & 1u) ? yv[r+1] * LSCALE : 0.0f;
      v.z = ((sel >> (r+2)) & 1u) ? yv[r+2] * LSCALE : 0.0f;
      v.w = ((sel >> (r+3)) & 1u) ? yv[r+3] * LSCALE : 0.0f;
      *(v4f*)(zr + r) = v;
    }
  }
}

// ---------------------------------------------------------------------------
// Kernel 2: out = z @ B^T  (K = 32 -> 8 WMMA steps, all A-operands in regs).
// z tile staged to LDS with gfx1250 async-to-LDS DMA (ASYNCcnt-tracked).
// Block = 16 tokens; 8 waves; wave w covers output tiles [w*32, w*32+32).
// ---------------------------------------------------------------------------
__global__ __launch_bounds__(256) void fly_expand(
    const float* __restrict__ z, const float* __restrict__ Bm,
    float* __restrict__ out)
{
  __shared__ __align__(16) float zs[16 * R_DIM];      // 2 KB z tile, shared by 8 waves

  const int tok0 = blockIdx.x * 16;

  if (threadIdx.x < 128) {                            // waves 0-3: no intra-wave divergence
    unsigned lds_off =
        (unsigned)(unsigned long long)(size_t)&zs[threadIdx.x * 4];
    unsigned long long ga =
        (unsigned long long)(size_t)(z + (size_t)tok0 * R_DIM + threadIdx.x * 4);
    asm volatile("global_load_async_to_lds_b128 %0, %1, off"
                 :: "v"(lds_off), "v"(ga) : "memory");
    asm volatile("s_wait_asynccnt 0" ::: "memory");
  }
  __syncthreads();

  const int lane  = threadIdx.x & 31;
  const int wv    = threadIdx.x >> 5;
  const int row   = lane & 15;
  const int khalf = (lane >> 4) * 2;
  const int thalf = (lane >> 4) * 8;

  // A-operands for all 8 K-steps, built once from LDS (ds_load_b64 pattern).
  v2f az[8];
  #pragma unroll
  for (int s = 0; s < 8; ++s)
    az[s] = *(const v2f*)(zs + row * R_DIM + 4 * s + khalf);

  for (int ot = wv * 32; ot < wv * 32 + 32; ++ot) {
    const int o0 = ot * 16;
    const float* bp = Bm + (size_t)(o0 + row) * R_DIM + khalf;  // B-operand: lane=N=o
    v8f c = {};
    #pragma unroll
    for (int s = 0; s < 8; ++s) {
      v2f b = *(const v2f*)(bp + 4 * s);
      c = __builtin_amdgcn_wmma_f32_16x16x4_f32(false, az[s], false, b, (short)0, c, false, false);
    }
    // Streaming 256 MB output: non-temporal stores, C/D layout row i / i+8.
    float* op = out + (size_t)(tok0 + thalf) * D_DIM + o0 + row;
    #pragma unroll
    for (int i = 0; i < 8; ++i)
      __builtin_nontemporal_store(c[i], op + (size_t)i * D_DIM);
  }
}

// ---------------------------------------------------------------------------
extern "C" void kernel_launch(void* const* d_in, const int* in_sizes, int n_in,
                              void* d_out, int out_size, void* d_ws, size_t ws_size,
                              hipStream_t stream) {
  (void)in_sizes; (void)n_in; (void)out_size; (void)ws_size;
  const float* x  = (const float*)d_in[0];
  const float* A  = (const float*)d_in[1];
  const float* Bm = (const float*)d_in[2];
  const float* d  = (const float*)d_in[3];
  float* out = (float*)d_out;
  float* z   = (float*)d_ws;          // 16384 * 32 f32 = 2 MB scratch

  fly_project_route<<<dim3(T_TOT / 16), dim3(256), 0, stream>>>(x, A, d, z);
  fly_expand       <<<dim3(T_TOT / 16), dim3(256), 0, stream>>>(z, Bm, out);
}